// GaussianHistogram_Dis_17343077941912
// MI455X (gfx1250) — compile-verified
//
#include <hip/hip_runtime.h>
#include <hip/hip_bf16.h>
#include <math.h>

// ---------------------------------------------------------------------------
// GaussianHistogram on MI455X (gfx1250).
//
// hist[k][c] = sum_n gauss(dist_n, center_k) * onehot(param_n, c)
// This is a (64 x 512K) @ (512K x 20) f32 GEMM -> mapped onto
// V_WMMA_F32_16X16X4_F32 (A: 16 bins x 4 bonds, B: 4 bonds x 16 classes,
// C: 16x16 f32 accumulator held in registers across the whole bond loop).
// Raw v_exp_f32 / v_sqrt_f32 via builtins: args are bounded, so the OCML
// range-fixup (cndmask + ldexp per exp) is dead weight in the hot loop.
// ---------------------------------------------------------------------------

typedef __attribute__((ext_vector_type(2))) float v2f;
typedef __attribute__((ext_vector_type(8))) float v8f;

#define N_BATCH   64
#define N_ATOMS   8000
#define N_BONDS   8000
#define N_BINS    64
#define N_CLS     20          // one_hot over 21 classes, column 0 dropped
#define CLS_PAD   32          // padded class dim in scratch (2 N-tiles of 16)
#define DELTA     0.0625f     // (4.0 - 0.0) / 64
#define INV_SIG   20.0f       // 1 / 0.05
#define GSCALE    0.49867785f // DELTA / (SIGMA * sqrt(2*pi))
#define NEGHL2E  -0.72134752f // -0.5 * log2(e)  (exp(-z^2/2) == exp2(NEGHL2E*z^2))

__global__ void zero_hist_kernel(float* __restrict__ p, int n) {
    int i = blockIdx.x * blockDim.x + threadIdx.x;
    if (i < n) p[i] = 0.0f;
}

__global__ void hist_wmma_kernel(const float* __restrict__ atoms,
                                 const int*   __restrict__ indices,
                                 float*       __restrict__ ws_hist) {
    const int lane   = threadIdx.x & 31;
    const int wave   = (blockIdx.x * blockDim.x + threadIdx.x) >> 5;
    const int nWaves = (gridDim.x * blockDim.x) >> 5;
    const int nGroups = (N_BATCH * N_BONDS) / 4;   // 4 bonds per WMMA K-step

    const int  row  = lane & 15;
    const bool hi   = lane >= 16;
    const int  slot = lane & 3;          // which of the 4 bonds this lane loads
    const int  selA = hi ? 2 : 0;        // A-matrix K slots per ISA VGPR layout
    const int  selB = hi ? 3 : 1;

    // Bin centers for my 4 M-tiles (constant across the loop).
    float centers[4];
#pragma unroll
    for (int m = 0; m < 4; ++m)
        centers[m] = DELTA * ((float)(m * 16 + row) + 0.5f);

    const int cls0 = row;        // N-tile 0: classes 0..15
    const int cls1 = 16 + row;   // N-tile 1: classes 16..31 (20..31 stay zero)

    v8f acc[4][2];
#pragma unroll
    for (int m = 0; m < 4; ++m)
#pragma unroll
        for (int n = 0; n < 2; ++n)
            acc[m][n] = (v8f){0.f, 0.f, 0.f, 0.f, 0.f, 0.f, 0.f, 0.f};

    for (int g = wave; g < nGroups; g += nWaves) {    // wave-uniform: EXEC all-1s
        const int bond0 = g * 4;                      // 8000 % 4 == 0 -> no batch straddle
        const int batch = bond0 / N_BONDS;
        const int myBond = bond0 + slot;

        const int* idx = indices + (size_t)myBond * 3;
        const int p  = idx[0];
        const int n1 = idx[1];
        const int n2 = idx[2];

        const float* ab = atoms + (size_t)batch * (N_ATOMS * 3);
        const float dx = ab[n1 * 3 + 0] - ab[n2 * 3 + 0];
        const float dy = ab[n1 * 3 + 1] - ab[n2 * 3 + 1];
        const float dz = ab[n1 * 3 + 2] - ab[n2 * 3 + 2];
        const float dist = __builtin_amdgcn_sqrtf(dx * dx + dy * dy + dz * dz);

        // Broadcast the two bond slots this lane-half needs (wave32 shuffle).
        const float dA = __shfl(dist, selA, 32);
        const float dB = __shfl(dist, selB, 32);
        const int   pA = __shfl(p,    selA, 32);
        const int   pB = __shfl(p,    selB, 32);

        // B tiles: one-hot over classes (4 x 16 per tile).
        v2f bm0, bm1;
        bm0.x = (pA == cls0) ? 1.0f : 0.0f;
        bm0.y = (pB == cls0) ? 1.0f : 0.0f;
        bm1.x = (pA == cls1) ? 1.0f : 0.0f;
        bm1.y = (pB == cls1) ? 1.0f : 0.0f;

#pragma unroll
        for (int m = 0; m < 4; ++m) {
            const float tA = (dA - centers[m]) * INV_SIG;
            const float tB = (dB - centers[m]) * INV_SIG;
            v2f am;
            // arg <= 0 always; hardware v_exp_f32 underflows to 0 like the ref
            am.x = GSCALE * __builtin_amdgcn_exp2f(tA * tA * NEGHL2E);
            am.y = GSCALE * __builtin_amdgcn_exp2f(tB * tB * NEGHL2E);
            acc[m][0] = __builtin_amdgcn_wmma_f32_16x16x4_f32(
                false, am, false, bm0, (short)0, acc[m][0], false, false);
            acc[m][1] = __builtin_amdgcn_wmma_f32_16x16x4_f32(
                false, am, false, bm1, (short)0, acc[m][1], false, false);
        }
    }

    // Flush register-resident tiles. C/D layout: VGPR r -> M = r + (hi?8:0),
    // N = lane&15. One flush per wave -> ~131K atomics total, negligible.
    const int binOff = hi ? 8 : 0;
#pragma unroll
    for (int m = 0; m < 4; ++m)
#pragma unroll
        for (int n = 0; n < 2; ++n)
#pragma unroll
            for (int r = 0; r < 8; ++r) {
                const int bin = m * 16 + r + binOff;
                const int cls = n * 16 + row;
                atomicAdd(&ws_hist[bin * CLS_PAD + cls], acc[m][n][r]);
            }
}

__global__ void normalize_kernel(const float* __restrict__ ws_hist,
                                 float* __restrict__ out) {
    __shared__ float denom[N_CLS];
    const int tid = threadIdx.x;
    if (tid < N_CLS) {
        float s = 0.0f;
        for (int k = 0; k < N_BINS; ++k)
            s += ws_hist[k * CLS_PAD + tid] + 1e-40f;   // ref adds 1e-40 per elem
        denom[tid] = s + 1e-20f;
    }
    __syncthreads();
    for (int i = tid; i < N_BINS * N_CLS; i += blockDim.x) {
        const int k = i / N_CLS;
        const int c = i % N_CLS;
        out[i] = (ws_hist[k * CLS_PAD + c] + 1e-40f) / denom[c];
    }
}

extern "C" void kernel_launch(void* const* d_in, const int* in_sizes, int n_in,
                              void* d_out, int out_size, void* d_ws, size_t ws_size,
                              hipStream_t stream) {
    const float* atoms   = (const float*)d_in[0];
    const int*   indices = (const int*)d_in[1];
    float*       out     = (float*)d_out;
    float*       hist    = (float*)d_ws;     // N_BINS * CLS_PAD f32 = 8 KB

    const int histElems = N_BINS * CLS_PAD;
    zero_hist_kernel<<<(histElems + 255) / 256, 256, 0, stream>>>(hist, histElems);
    hist_wmma_kernel<<<256, 256, 0, stream>>>(atoms, indices, hist);
    normalize_kernel<<<1, 256, 0, stream>>>(hist, out);
}